// PositionalEncoding2D_3418793967837
// MI455X (gfx1250) — compile-verified
//
#include <hip/hip_runtime.h>

// ---------------------------------------------------------------------------
// PositionalEncoding2D: out[b,s,:] = x[b,s,:] + concat(pe_h[cy], pe_w[cx])
// Pure bandwidth kernel: ~257 MiB HBM traffic -> ~11.5 us floor @ 23.3 TB/s.
// PE tables (400 KB) stay L2/WGP$-resident; x/out streamed with NT hints.
// ---------------------------------------------------------------------------

typedef float v4f __attribute__((ext_vector_type(4)));

constexpr int D_MODEL        = 1024;
constexpr int HALF_D         = 512;
constexpr int BATCH          = 8;
constexpr int SEQ            = 4096;
constexpr int N_ROWS         = BATCH * SEQ;   // 32768
constexpr int ROWS_PER_BLOCK = 4;             // 4 independent mem chains / thread
constexpr int THREADS        = 256;           // 8 wave32 waves, 1 row = 256 float4

__global__ __launch_bounds__(THREADS)
void pe2d_add_kernel(const float* __restrict__ x,
                     const float* __restrict__ boxes,
                     const float* __restrict__ pe_h,
                     const float* __restrict__ pe_w,
                     float* __restrict__ out)
{
    const int row0 = blockIdx.x * ROWS_PER_BLOCK;
    const int tid  = threadIdx.x;
    const int d    = tid * 4;                  // float offset within the row
    const bool first_half = (d < HALF_D);

#pragma unroll
    for (int r = 0; r < ROWS_PER_BLOCK; ++r) {
        const int row = row0 + r;
        const size_t off = (size_t)row * D_MODEL + d;

        // gfx1250 prefetch of next row's x chunk (global_prefetch_b8, no cnt)
        if (r + 1 < ROWS_PER_BLOCK) {
            __builtin_prefetch(x + off + D_MODEL, 0, 0);
        }

        // Uniform (per-row) box coords -> scalar-path load. Exact reference
        // arithmetic: (b0+b2)*0.5*99 with trunc-toward-zero, clamp [0,99].
        const v4f bx = *reinterpret_cast<const v4f*>(boxes + (size_t)row * 4);
        int cx = (int)((bx.x + bx.z) * 0.5f * 99.0f);
        int cy = (int)((bx.y + bx.w) * 0.5f * 99.0f);
        cx = cx < 0 ? 0 : (cx > 99 ? 99 : cx);
        cy = cy < 0 ? 0 : (cy > 99 ? 99 : cy);

        // PE gather: lanes 0..127 read pe_h[cy] contiguously, lanes 128..255
        // read pe_w[cx] contiguously. Default (RT) policy keeps the 400 KB
        // tables cache-resident.
        const float* pe_ptr = first_half
            ? (pe_h + (size_t)cy * HALF_D + d)
            : (pe_w + (size_t)cx * HALF_D + (d - HALF_D));
        const v4f pe = *reinterpret_cast<const v4f*>(pe_ptr);

        // Streamed x/out: b128 with non-temporal hint so the 256 MiB one-pass
        // stream doesn't evict the hot PE tables.
        const v4f xv = __builtin_nontemporal_load(
            reinterpret_cast<const v4f*>(x + off));
        __builtin_nontemporal_store(xv + pe,
            reinterpret_cast<v4f*>(out + off));
    }
}

extern "C" void kernel_launch(void* const* d_in, const int* in_sizes, int n_in,
                              void* d_out, int out_size, void* d_ws, size_t ws_size,
                              hipStream_t stream)
{
    const float* x     = (const float*)d_in[0];
    const float* boxes = (const float*)d_in[1];
    const float* pe_h  = (const float*)d_in[2];
    const float* pe_w  = (const float*)d_in[3];
    float* out         = (float*)d_out;

    dim3 grid(N_ROWS / ROWS_PER_BLOCK);   // 8192 blocks
    dim3 block(THREADS);
    hipLaunchKernelGGL(pe2d_add_kernel, grid, block, 0, stream,
                       x, boxes, pe_h, pe_w, out);
}